// TopK_82437602279881
// MI455X (gfx1250) — compile-verified
//
#include <hip/hip_runtime.h>
#include <hip/hip_bf16.h>

// ---------------------------------------------------------------------------
// TopK SAE forward for MI455X (gfx1250).
//   0. init            : zero col_active / accumulators
//   1. row_stats       : row stats for x/y; writes xn = (x-mu)*rstd - b_dec
//   2. enc_gemm        : dense acts = relu(xn @ W_enc) via V_WMMA_F32_16X16X4,
//                        tiles staged by TENSOR_LOAD_TO_LDS (TDM, double-buffered,
//                        LDS padding via descriptor to kill bank conflicts)
//   3. topk32          : per-row radix-select top-32, mark col_active
//   4. dead_features   : nb update, dead flags, num_dead, dead_any
//   5. decode_aux      : sparsify acts row, y_pred (+out), losses, masked
//                        top-512 aux decode
//   6. finalize        : assemble 7 scalar outputs
// ---------------------------------------------------------------------------

typedef __attribute__((ext_vector_type(2))) float    v2f;
typedef __attribute__((ext_vector_type(8))) float    v8f;
typedef __attribute__((ext_vector_type(4))) unsigned v4u;
typedef __attribute__((ext_vector_type(4))) int      v4i;
typedef __attribute__((ext_vector_type(8))) int      v8i;

#define BATCH   2048
#define D_IN    768
#define D_OUT   768
#define DICT    24576
#define TOPK    32
#define TOPKAUX 512

#if defined(__has_builtin)
#if __has_builtin(__builtin_amdgcn_tensor_load_to_lds) && \
    __has_builtin(__builtin_amdgcn_s_wait_tensorcnt)
#define USE_TDM 1
#endif
#endif

// ---------------------------------------------------------------- init ------
__global__ void k_init(int* colact, float* accF, int* accI) {
  int i = blockIdx.x * blockDim.x + threadIdx.x;
  if (i < DICT) colact[i] = 0;
  if (blockIdx.x == 0) {
    if (threadIdx.x < 4) accF[threadIdx.x] = 0.0f;
    if (threadIdx.x < 2) accI[threadIdx.x] = 0;
  }
}

// ----------------------------------------------------------- row stats ------
__global__ void k_row_stats(const float* __restrict__ x,
                            const float* __restrict__ y,
                            const float* __restrict__ bdec,
                            float* __restrict__ xn,
                            float* ym, float* ys, float* yr) {
  __shared__ float r1[256], r2[256], sstat[2];
  int row = blockIdx.x, tid = threadIdx.x;

  // x stats
  float s = 0.f, s2 = 0.f;
  for (int c = tid; c < D_IN; c += 256) {
    float v = x[row * D_IN + c];
    s += v; s2 += v * v;
  }
  r1[tid] = s; r2[tid] = s2; __syncthreads();
  for (int o = 128; o > 0; o >>= 1) {
    if (tid < o) { r1[tid] += r1[tid + o]; r2[tid] += r2[tid + o]; }
    __syncthreads();
  }
  if (tid == 0) {
    float mean = r1[0] / (float)D_IN;
    float var  = (r2[0] - (float)D_IN * mean * mean) / (float)(D_IN - 1);
    float sd   = sqrtf(fmaxf(var, 0.f));
    sstat[0] = mean; sstat[1] = 1.0f / (sd + 1e-5f);
  }
  __syncthreads();
  float mean = sstat[0], rstd = sstat[1];
  for (int c = tid; c < D_IN; c += 256)
    xn[row * D_IN + c] = (x[row * D_IN + c] - mean) * rstd - bdec[c];
  __syncthreads();

  // y stats
  s = 0.f; s2 = 0.f;
  for (int c = tid; c < D_OUT; c += 256) {
    float v = y[row * D_OUT + c];
    s += v; s2 += v * v;
  }
  r1[tid] = s; r2[tid] = s2; __syncthreads();
  for (int o = 128; o > 0; o >>= 1) {
    if (tid < o) { r1[tid] += r1[tid + o]; r2[tid] += r2[tid + o]; }
    __syncthreads();
  }
  if (tid == 0) {
    float m = r1[0] / (float)D_OUT;
    float var = (r2[0] - (float)D_OUT * m * m) / (float)(D_OUT - 1);
    float sd  = sqrtf(fmaxf(var, 0.f));
    ym[row] = m; ys[row] = sd; yr[row] = 1.0f / (sd + 1e-5f);
  }
}

// ------------------------------------------------------- encoder GEMM -------
// Block 256 (8 waves = 4m x 2n). Tile M=64, N=128, K-chunk 32.
// LDS A row stride 36, B row stride 144 (TDM-inserted padding):
//   A frag b64 loads: half-waves hit disjoint mod-4 bank residues -> no conflict
//   B frag b32 loads: half-waves separated by 32 banks -> no conflict
#define KCH  32
#define APAD 36
#define BPAD 144
#define NKCH (D_IN / KCH)

#ifdef USE_TDM
__device__ __attribute__((always_inline)) inline unsigned lds_byte_addr(const void* p) {
  // generic pointer to LDS: low 32 bits of the flat address are the LDS offset
  return (unsigned)(unsigned long long)p;
}

// 2D tile DMA: global (row-major, stride0 elems/row) -> LDS (+optional padding)
__device__ __attribute__((always_inline)) inline void
tdm_load_2d(unsigned lds_addr, const void* gptr, unsigned g1w0,
            unsigned tdim0, unsigned tdim1, unsigned tile0, unsigned tile1,
            unsigned long long stride0) {
  unsigned long long ga = (unsigned long long)gptr;
  v4u g0;
  g0[0] = 1u;                                          // count=1, user mode
  g0[1] = lds_addr;                                    // lds_addr
  g0[2] = (unsigned)(ga & 0xFFFFFFFFu);                // global_addr lo
  g0[3] = (unsigned)((ga >> 32) & 0x1FFFFFFu) | (2u << 30);  // addr hi | type=2
  v8i g1;
  g1[0] = (int)g1w0;                                   // mask=0|data_size|pad cfg
  g1[1] = (int)((tdim0 & 0xFFFFu) << 16);              // tensor_dim0[15:0]
  g1[2] = (int)(((tdim0 >> 16) & 0xFFFFu) | ((tdim1 & 0xFFFFu) << 16));
  g1[3] = (int)(((tdim1 >> 16) & 0xFFFFu) | ((tile0 & 0xFFFFu) << 16));
  g1[4] = (int)(tile1 & 0xFFFFu);                      // tile_dim1 (tile_dim2=0)
  g1[5] = (int)(unsigned)(stride0 & 0xFFFFFFFFu);      // dim0_stride lo
  g1[6] = (int)(unsigned)((stride0 >> 32) & 0xFFFFu);  // dim0_stride hi
  g1[7] = 0;
  v4i z4 = {0, 0, 0, 0};
#if defined(__clang_major__) && (__clang_major__ >= 23)
  v8i z8 = {0, 0, 0, 0, 0, 0, 0, 0};
  __builtin_amdgcn_tensor_load_to_lds(g0, g1, z4, z4, z8, 0);
#else
  __builtin_amdgcn_tensor_load_to_lds(g0, g1, z4, z4, 0);
#endif
}
#endif  // USE_TDM

__global__ void k_enc_gemm(const float* __restrict__ xn,
                           const float* __restrict__ Wenc,
                           float* __restrict__ acts) {
  __shared__ float As[2][64 * APAD];
  __shared__ float Bs[2][KCH * BPAD];

  const int tid   = threadIdx.x;
  const int lane  = tid & 31;
  const int wid   = tid >> 5;
  const int wm    = wid & 3;        // wave m-tile 0..3
  const int wn    = wid >> 2;       // wave n-strip 0..1
  const int lrow  = lane & 15;
  const int lhalf = lane >> 4;      // 0: K pair {0,1}, 1: K pair {2,3}

  const int m0 = blockIdx.y * 64;
  const int n0 = blockIdx.x * 128;

  v8f acc[4];
#pragma unroll
  for (int t = 0; t < 4; ++t)
#pragma unroll
    for (int r = 0; r < 8; ++r) acc[t][r] = 0.0f;

#ifdef USE_TDM
  // data_size=4B, pad_enable, A: pad 4 DWORDs every 32; B: pad 16 every 128
  const unsigned AFLAGS = (2u << 16) | (1u << 20) | (4u << 22) | (3u << 25);
  const unsigned BFLAGS = (2u << 16) | (1u << 20) | (6u << 22) | (15u << 25);

  if (wid == 0) {
    tdm_load_2d(lds_byte_addr(&As[0][0]), xn + (size_t)m0 * D_IN,
                AFLAGS, KCH, 64, KCH, 64, D_IN);
    tdm_load_2d(lds_byte_addr(&Bs[0][0]), Wenc + (size_t)n0,
                BFLAGS, 128, KCH, 128, KCH, DICT);
  }
  for (int i = 0; i < NKCH; ++i) {
    const int buf = i & 1;
    if (wid == 0) {
      if (i + 1 < NKCH) {
        const int kc = (i + 1) * KCH;
        tdm_load_2d(lds_byte_addr(&As[buf ^ 1][0]),
                    xn + (size_t)m0 * D_IN + kc,
                    AFLAGS, KCH, 64, KCH, 64, D_IN);
        tdm_load_2d(lds_byte_addr(&Bs[buf ^ 1][0]),
                    Wenc + (size_t)kc * DICT + n0,
                    BFLAGS, 128, KCH, 128, KCH, DICT);
        __builtin_amdgcn_s_wait_tensorcnt(2);   // current buffer's 2 DMAs done
      } else {
        __builtin_amdgcn_s_wait_tensorcnt(0);
      }
    }
    __syncthreads();
#else
  for (int i = 0; i < NKCH; ++i) {
    const int buf = 0;
    const int kc0 = i * KCH;
#pragma unroll
    for (int j = 0; j < 8; ++j) {          // A: 64 rows x 32 k
      int e = tid + j * 256;
      int rr = e >> 5, kk = e & 31;
      As[0][rr * APAD + kk] = xn[(size_t)(m0 + rr) * D_IN + kc0 + kk];
    }
#pragma unroll
    for (int j = 0; j < 16; ++j) {         // B: 32 k x 128 n
      int e = tid + j * 256;
      int kk = e >> 7, nn = e & 127;
      Bs[0][kk * BPAD + nn] = Wenc[(size_t)(kc0 + kk) * DICT + (n0 + nn)];
    }
    __syncthreads();
#endif

#pragma unroll
    for (int kk = 0; kk < KCH; kk += 4) {
      v2f a;
      const int arow = (wm * 16 + lrow) * APAD + kk + 2 * lhalf;
      a.x = As[buf][arow];
      a.y = As[buf][arow + 1];
#pragma unroll
      for (int nt = 0; nt < 4; ++nt) {
        const int ncol = wn * 64 + nt * 16 + lrow;
        v2f b;
        b.x = Bs[buf][(kk + 2 * lhalf) * BPAD + ncol];
        b.y = Bs[buf][(kk + 2 * lhalf + 1) * BPAD + ncol];
        acc[nt] = __builtin_amdgcn_wmma_f32_16x16x4_f32(
            false, a, false, b, (short)0, acc[nt], false, false);
      }
    }
    __syncthreads();
  }

  // epilogue: relu + store dense acts (scratch region of d_out)
#pragma unroll
  for (int nt = 0; nt < 4; ++nt) {
#pragma unroll
    for (int r = 0; r < 8; ++r) {
      int grow = m0 + wm * 16 + (lhalf ? r + 8 : r);
      int gcol = n0 + wn * 64 + nt * 16 + lrow;
      acts[(size_t)grow * DICT + gcol] = fmaxf(acc[nt][r], 0.0f);
    }
  }
}

// --------------------------------------------------- radix select (LDS) -----
__device__ inline unsigned int sel_key(float v, unsigned char d, int masked) {
  unsigned int b = __float_as_uint(v);            // v >= 0 -> monotonic
  return masked ? (d ? (b + 1u) : 0u) : b;
}

// Finds threshold T (key space): count(key > T) < K <= count(key >= T).
// krem = K - count(key > T).
__device__ void radix_select(const float* v, const unsigned char* dflag,
                             int n, int K, unsigned int* hist, int* st,
                             int masked, unsigned int* T_out, int* krem_out) {
  int tid = threadIdx.x;
  unsigned int prefix = 0;
  int krem = K;
  for (int shift = 24; shift >= 0; shift -= 8) {
    for (int b = tid; b < 256; b += blockDim.x) hist[b] = 0;
    __syncthreads();
    for (int i = tid; i < n; i += blockDim.x) {
      unsigned int key = sel_key(v[i], masked ? dflag[i] : 0, masked);
      bool ok = (shift == 24) ||
                ((key >> (shift + 8)) == (prefix >> (shift + 8)));
      if (ok) atomicAdd(&hist[(key >> shift) & 0xFFu], 1u);
    }
    __syncthreads();
    if (tid == 0) {
      unsigned int cum = 0; int chosen = 0; int newk = krem;
      for (int b = 255; b >= 0; --b) {
        unsigned int h = hist[b];
        if (cum + h >= (unsigned)krem) { chosen = b; newk = krem - (int)cum; break; }
        cum += h;
      }
      st[0] = chosen; st[1] = newk;
    }
    __syncthreads();
    prefix |= ((unsigned)st[0]) << shift;
    krem = st[1];
    __syncthreads();
  }
  *T_out = prefix; *krem_out = krem;
}

// ----------------------------------------------------------- top-32 ---------
__global__ void k_topk32(const float* __restrict__ acts,
                         float* __restrict__ tv, int* __restrict__ tix,
                         int* __restrict__ colact) {
  extern __shared__ char smem[];
  float*        rowv = (float*)smem;                 // DICT
  unsigned int* hist = (unsigned int*)(rowv + DICT); // 256
  int*          st   = (int*)(hist + 256);          // 8

  int row = blockIdx.x, tid = threadIdx.x;
  for (int i = tid; i < DICT; i += 256) rowv[i] = acts[(size_t)row * DICT + i];
  __syncthreads();

  unsigned int T; int krem;
  radix_select(rowv, nullptr, DICT, TOPK, hist, st, 0, &T, &krem);

  if (tid == 0) { st[2] = 0; st[3] = 0; }
  if (tid < TOPK) { tv[row * TOPK + tid] = 0.f; tix[row * TOPK + tid] = tid; }
  __syncthreads();

  for (int i = tid; i < DICT; i += 256) {
    unsigned int key = __float_as_uint(rowv[i]);
    if (key > T) {
      int s = atomicAdd(&st[2], 1);
      if (s < TOPK) {
        tv[row * TOPK + s] = rowv[i];
        tix[row * TOPK + s] = i;
        colact[i] = 1;                 // v > T >= 0  ->  v > 0
      }
    }
  }
  __syncthreads();
  if (T > 0) {
    int base = st[2];
    for (int i = tid; i < DICT; i += 256) {
      unsigned int key = __float_as_uint(rowv[i]);
      if (key == T) {
        int e = atomicAdd(&st[3], 1);
        if (e < krem) {
          int s = base + e;
          if (s < TOPK) {
            tv[row * TOPK + s] = rowv[i];
            tix[row * TOPK + s] = i;
            colact[i] = 1;
          }
        }
      }
    }
  }
}

// ------------------------------------------------------ dead features -------
__global__ void k_dead(const int* __restrict__ colact,
                       const float* __restrict__ nb,
                       int* __restrict__ dead, int* accI) {
  int i = blockIdx.x * blockDim.x + threadIdx.x;
  if (i >= DICT) return;
  float nbn = colact[i] ? 0.0f : (nb[i] + 1.0f);
  int d = (nbn >= 100.0f) ? 1 : 0;
  dead[i] = d;
  if (d) atomicOr(&accI[0], 1);
  if (nbn > 100.0f) atomicAdd(&accI[1], 1);
}

// ------------------------------------------- decode + aux + losses ----------
__global__ void k_decode_aux(float* __restrict__ acts,         // dense in, sparse out
                             const float* __restrict__ yt,
                             const float* __restrict__ Wdec,
                             const float* __restrict__ bdec,
                             const float* __restrict__ ym,
                             const float* __restrict__ ys,
                             const float* __restrict__ yr,
                             const float* __restrict__ tv,
                             const int* __restrict__ tix,
                             const int* __restrict__ dead,
                             float* __restrict__ yout,
                             float* accF) {
  extern __shared__ char smem[];
  float*         rowv  = (float*)smem;                       // DICT floats
  unsigned char* dflag = (unsigned char*)(rowv + DICT);      // DICT bytes
  unsigned int*  hist  = (unsigned int*)(dflag + DICT);      // 256
  int*           st    = (int*)(hist + 256);                 // 8
  float*         sv    = (float*)(st + 8);                   // 512
  int*           si    = (int*)(sv + TOPKAUX);               // 512
  float*         tvl   = (float*)(si + TOPKAUX);             // 32
  int*           til   = (int*)(tvl + TOPK);                 // 32
  float*         red   = (float*)(til + TOPK);               // 256

  int row = blockIdx.x, tid = threadIdx.x;

  for (int i = tid; i < DICT; i += 256) {
    rowv[i]  = acts[(size_t)row * DICT + i];
    dflag[i] = (unsigned char)dead[i];
  }
  if (tid < TOPK) { tvl[tid] = tv[row * TOPK + tid]; til[tid] = tix[row * TOPK + tid]; }
  __syncthreads();

  // sparsify output row: zero then scatter positive top-k values
  for (int i = tid; i < DICT; i += 256) acts[(size_t)row * DICT + i] = 0.0f;
  __syncthreads();
  if (tid < TOPK) {
    float v = tvl[tid];
    if (v > 0.0f) acts[(size_t)row * DICT + til[tid]] = v;
  }

  // y_pred = sum_k v * W_dec[idx,:] + b_dec   (3 cols per thread, strided)
  float yp[3], res[3];
#pragma unroll
  for (int j = 0; j < 3; ++j) yp[j] = bdec[tid + j * 256];
  for (int t = 0; t < TOPK; ++t) {
    float v = tvl[t];
    const float* wrow = Wdec + (size_t)til[t] * D_OUT;
#pragma unroll
    for (int j = 0; j < 3; ++j) yp[j] += v * wrow[tid + j * 256];
  }
  float mean = ym[row], sd = ys[row], rs = yr[row];
  float l2p = 0.0f;
#pragma unroll
  for (int j = 0; j < 3; ++j) {
    int c = tid + j * 256;
    float yv = yt[row * D_OUT + c];
    float yn = (yv - mean) * rs;
    float d  = yp[j] - yn;
    l2p += d * d;
    res[j] = yn - yp[j];
    yout[row * D_OUT + c] = yp[j] * sd + mean;   // de-normalized prediction
  }

  red[tid] = l2p; __syncthreads();
  for (int o = 128; o > 0; o >>= 1) {
    if (tid < o) red[tid] += red[tid + o];
    __syncthreads();
  }
  if (tid == 0) {
    atomicAdd(&accF[0], red[0]);                  // l2 SSE
    float s1 = 0.f, c0 = 0.f;
    for (int t = 0; t < TOPK; ++t) { s1 += tvl[t]; if (tvl[t] > 0.f) c0 += 1.f; }
    atomicAdd(&accF[1], s1);                      // l1 sum
    atomicAdd(&accF[2], c0);                      // l0 count
  }
  __syncthreads();

  // masked top-512 over dead columns
  unsigned int T; int krem;
  radix_select(rowv, dflag, DICT, TOPKAUX, hist, st, 1, &T, &krem);
  if (tid == 0) { st[2] = 0; st[3] = 0; }
  __syncthreads();
  for (int i = tid; i < DICT; i += 256) {
    unsigned int key = sel_key(rowv[i], dflag[i], 1);
    if (key > T) {
      int s = atomicAdd(&st[2], 1);
      if (s < TOPKAUX) { sv[s] = rowv[i]; si[s] = i; }
    }
  }
  __syncthreads();
  int base = st[2] < TOPKAUX ? st[2] : TOPKAUX;
  if (T > 0) {
    for (int i = tid; i < DICT; i += 256) {
      unsigned int key = sel_key(rowv[i], dflag[i], 1);
      if (key == T) {
        int e = atomicAdd(&st[3], 1);
        if (e < krem) {
          int s = base + e;
          if (s < TOPKAUX) { sv[s] = rowv[i]; si[s] = i; }
        }
      }
    }
  }
  __syncthreads();
  int taken_eq = 0;
  if (T > 0) taken_eq = (st[3] < krem) ? st[3] : krem;
  int nsel = base + taken_eq;
  if (nsel > TOPKAUX) nsel = TOPKAUX;

  // y_pred_aux and aux SSE
  float ya[3] = {0.f, 0.f, 0.f};
  for (int t = 0; t < nsel; ++t) {
    float v = sv[t];
    const float* wrow = Wdec + (size_t)si[t] * D_OUT;
#pragma unroll
    for (int j = 0; j < 3; ++j) ya[j] += v * wrow[tid + j * 256];
  }
  float auxp = 0.f;
#pragma unroll
  for (int j = 0; j < 3; ++j) {
    float d = ya[j] - res[j];
    auxp += d * d;
  }
  red[tid] = auxp; __syncthreads();
  for (int o = 128; o > 0; o >>= 1) {
    if (tid < o) red[tid] += red[tid + o];
    __syncthreads();
  }
  if (tid == 0) atomicAdd(&accF[3], red[0]);      // aux SSE
}

// --------------------------------------------------------- finalize ---------
__global__ void k_finalize(const float* accF, const int* accI, float* out7) {
  if (threadIdx.x == 0 && blockIdx.x == 0) {
    const float nel = (float)BATCH * (float)D_OUT;
    float l2   = accF[0] / nel;
    float l1n  = accF[1] / (float)BATCH;
    float l1l  = 0.001f * l1n;
    float l0   = accF[2] / (float)BATCH;
    float aux  = accI[0] ? (0.03125f * accF[3] / nel) : 0.0f;
    out7[0] = l2 + l1l + aux;   // loss
    out7[1] = l2;               // l2_loss
    out7[2] = l1l;              // l1_loss
    out7[3] = l0;               // l0_norm
    out7[4] = l1n;              // l1_norm
    out7[5] = aux;              // aux_loss
    out7[6] = (float)accI[1];   // num_dead_features
  }
}

// ------------------------------------------------------------ launch --------
extern "C" void kernel_launch(void* const* d_in, const int* in_sizes, int n_in,
                              void* d_out, int out_size, void* d_ws, size_t ws_size,
                              hipStream_t stream) {
  const float* x_in  = (const float*)d_in[0];
  const float* y_tgt = (const float*)d_in[1];
  const float* nb    = (const float*)d_in[2];
  const float* Wenc  = (const float*)d_in[3];
  const float* Wdec  = (const float*)d_in[4];
  const float* bdec  = (const float*)d_in[5];

  float* out  = (float*)d_out;
  float* yout = out;                                   // [B, D_OUT]
  float* acts = out + (size_t)BATCH * D_OUT;           // [B, DICT] (scratch->sparse)
  float* scal = acts + (size_t)BATCH * DICT;           // 7 scalars

  // workspace layout
  float* wsf = (float*)d_ws;
  float* xn  = wsf;                                    // BATCH * D_IN (normalized)
  float* ym  = xn + (size_t)BATCH * D_IN;
  float* ys  = ym + BATCH;
  float* yr  = ys + BATCH;
  float* tv  = yr + BATCH;                             // BATCH*TOPK
  int*   tix = (int*)(tv + (size_t)BATCH * TOPK);      // BATCH*TOPK
  int*   colact = tix + (size_t)BATCH * TOPK;          // DICT
  int*   dead   = colact + DICT;                       // DICT
  float* accF   = (float*)(dead + DICT);               // 4
  int*   accI   = (int*)(accF + 4);                    // 2

  const size_t topk_smem = (size_t)DICT * 4 + 256 * 4 + 8 * 4;
  const size_t dec_smem  = (size_t)DICT * 4 + DICT + 256 * 4 + 8 * 4 +
                           TOPKAUX * 4 + TOPKAUX * 4 + TOPK * 4 + TOPK * 4 +
                           256 * 4;

  k_init<<<dim3((DICT + 255) / 256), 256, 0, stream>>>(colact, accF, accI);
  k_row_stats<<<dim3(BATCH), 256, 0, stream>>>(x_in, y_tgt, bdec, xn, ym, ys, yr);
  k_enc_gemm<<<dim3(DICT / 128, BATCH / 64), 256, 0, stream>>>(xn, Wenc, acts);
  k_topk32<<<dim3(BATCH), 256, topk_smem, stream>>>(acts, tv, tix, colact);
  k_dead<<<dim3((DICT + 255) / 256), 256, 0, stream>>>(colact, nb, dead, accI);
  k_decode_aux<<<dim3(BATCH), 256, dec_smem, stream>>>(
      acts, y_tgt, Wdec, bdec, ym, ys, yr, tv, tix, dead, yout, accF);
  k_finalize<<<dim3(1), 32, 0, stream>>>(accF, accI, scal);
}